// SAGE_11072425690020
// MI455X (gfx1250) — compile-verified
//
#include <hip/hip_runtime.h>
#include <stdint.h>

typedef __attribute__((ext_vector_type(2))) float v2f;
typedef __attribute__((ext_vector_type(8))) float v8f;
typedef __attribute__((ext_vector_type(4))) int   v4i;

typedef __attribute__((address_space(1))) v4i g_v4i;   // global int4
typedef __attribute__((address_space(3))) v4i l_v4i;   // LDS int4

#define FANOUT 10

// ---------------------------------------------------------------------------
// gfx1250 async copy: global -> LDS, 16B per lane, tracked on ASYNCcnt.
// ---------------------------------------------------------------------------
__device__ __forceinline__ void async_ld_b128(const float* g, uint32_t lds_off) {
#if __has_builtin(__builtin_amdgcn_global_load_async_to_lds_b128)
  __builtin_amdgcn_global_load_async_to_lds_b128(
      (g_v4i*)(uintptr_t)g, (l_v4i*)(uintptr_t)lds_off, 0, 0);
#else
  asm volatile("global_load_async_to_lds_b128 %0, %1, off"
               :: "v"(lds_off), "v"(g) : "memory");
#endif
}

__device__ __forceinline__ void wait_async0() {
#if __has_builtin(__builtin_amdgcn_s_wait_asynccnt)
  __builtin_amdgcn_s_wait_asynccnt(0);
#else
  asm volatile("s_wait_asynccnt 0" ::: "memory");
#endif
}

// ---------------------------------------------------------------------------
// Kernel 1: neighbor aggregation (mean over contiguous groups of 10 edges).
// grid = n_tgt blocks, block = din threads. Each src-row read is a fully
// coalesced din*4-byte burst across the block; 10 independent loads in flight.
// ---------------------------------------------------------------------------
__global__ __launch_bounds__(256) void sage_agg(const float* __restrict__ x,
                                                const int* __restrict__ src,
                                                float* __restrict__ agg,
                                                int din) {
  const int t = blockIdx.x;
  const int d = threadIdx.x;                 // blockDim.x == din
  const long base = (long)t * FANOUT;
  float s = 0.0f;
#pragma unroll
  for (int j = 0; j < FANOUT; ++j) {
    const int r = src[base + j];
    s += x[(long)r * din + d];
  }
  agg[(long)t * din + d] = s * (1.0f / FANOUT);
}

// ---------------------------------------------------------------------------
// Kernel 2: fused dual-GEMM  out = agg @ Wl^T + xt @ Wr^T + b  via
// V_WMMA_F32_16X16X4_F32, 2x2 register blocking (32x32 per wave, 4
// independent accumulator chains).
//
// Block = NW waves covering 32 rows x NW*32 cols (== dout). The 32xK A and X
// tiles are shared by all waves -> staged once into LDS with the async copy
// engine (ASYNCcnt), then fed via conflict-free ds_load_b64 (row stride K+4).
// Only the per-wave weight fragments are loaded from global in the K-loop:
// per k-step 4 ds_load_b64 + 4 global_load_b64 feed 8 WMMAs.
//
// Fragment layouts (CDNA5 ISA §7.12.2):
//   A 16x4 :  lane L -> M = L&15, holds K = {kb, kb+1}, kb = (L>>4)*2
//   B 4x16 :  lane L -> N = L&15, holds K = {kb, kb+1}   (B[k][n] = W[n][k])
//   C/D    :  lane L -> N = L&15, VGPR v -> M = v + 8*(L>>4)
// ---------------------------------------------------------------------------
template <int RELU, int K, int NW>
__global__ __launch_bounds__(NW * 32) void sage_gemm_wmma(
    const float* __restrict__ A,    // agg   : n_rows x K
    const float* __restrict__ X,    // x_tgt : n_rows x K
    const float* __restrict__ Wl,   // dout x K (row-major)
    const float* __restrict__ Wr,   // dout x K
    const float* __restrict__ bias, // dout
    float* __restrict__ out,        // n_rows x dout
    int dout) {
  constexpr int NT  = NW * 32;     // threads per block
  constexpr int LK  = K + 4;       // padded LDS row stride (floats)
  constexpr int CPR = K / 4;       // 16B chunks per row

  __shared__ float sA[32 * LK];
  __shared__ float sX[32 * LK];

  const int tid  = threadIdx.x;
  const int wave = tid >> 5;
  const int lane = tid & 31;
  const int base_m = blockIdx.x * 32;
  const int base_n = wave * 32;

  const int col = lane & 15;
  const int hi  = lane >> 4;
  const int kb0 = hi * 2;

  // ---- async stage of the block-shared 32xK A and X tiles into LDS ----
  {
    const float* gA = A + (size_t)base_m * K;    // 32*K contiguous floats
    const float* gX = X + (size_t)base_m * K;
    const uint32_t lA = (uint32_t)(uintptr_t)sA; // LDS byte offset = addr[31:0]
    const uint32_t lX = (uint32_t)(uintptr_t)sX;
    for (int i = tid; i < 32 * CPR; i += NT) {
      const int r = i / CPR;
      const int c = i - r * CPR;
      const uint32_t loff = (uint32_t)(r * LK + c * 4) * 4u;
      const size_t goff = (size_t)r * K + (size_t)c * 4;
      async_ld_b128(gA + goff, lA + loff);
      async_ld_b128(gX + goff, lX + loff);
    }
  }
  wait_async0();
  __syncthreads();

  const float* wl0 = Wl + (size_t)(base_n + col) * K;
  const float* wl1 = wl0 + (size_t)16 * K;
  const float* wr0 = Wr + (size_t)(base_n + col) * K;
  const float* wr1 = wr0 + (size_t)16 * K;

  const float bv0 = bias[base_n + col];
  const float bv1 = bias[base_n + 16 + col];
  v8f acc00 = {bv0, bv0, bv0, bv0, bv0, bv0, bv0, bv0};
  v8f acc01 = {bv1, bv1, bv1, bv1, bv1, bv1, bv1, bv1};
  v8f acc10 = acc00;
  v8f acc11 = acc01;

  const float* sArow0 = sA + col * LK;          // M rows 0..15
  const float* sArow1 = sA + (col + 16) * LK;   // M rows 16..31
  const float* sXrow0 = sX + col * LK;
  const float* sXrow1 = sX + (col + 16) * LK;

  for (int k0 = 0; k0 < K; k0 += 4) {
    const int kb = k0 + kb0;
    // --- agg @ Wl^T path ---
    const v2f a0 = *(const v2f*)(sArow0 + kb);
    const v2f a1 = *(const v2f*)(sArow1 + kb);
    const v2f l0 = *(const v2f*)(wl0 + kb);
    const v2f l1 = *(const v2f*)(wl1 + kb);
    acc00 = __builtin_amdgcn_wmma_f32_16x16x4_f32(false, a0, false, l0,
                                                  (short)0, acc00, false, false);
    acc01 = __builtin_amdgcn_wmma_f32_16x16x4_f32(false, a0, false, l1,
                                                  (short)0, acc01, false, false);
    acc10 = __builtin_amdgcn_wmma_f32_16x16x4_f32(false, a1, false, l0,
                                                  (short)0, acc10, false, false);
    acc11 = __builtin_amdgcn_wmma_f32_16x16x4_f32(false, a1, false, l1,
                                                  (short)0, acc11, false, false);
    // --- x_tgt @ Wr^T path ---
    const v2f x0 = *(const v2f*)(sXrow0 + kb);
    const v2f x1 = *(const v2f*)(sXrow1 + kb);
    const v2f r0 = *(const v2f*)(wr0 + kb);
    const v2f r1 = *(const v2f*)(wr1 + kb);
    acc00 = __builtin_amdgcn_wmma_f32_16x16x4_f32(false, x0, false, r0,
                                                  (short)0, acc00, false, false);
    acc01 = __builtin_amdgcn_wmma_f32_16x16x4_f32(false, x0, false, r1,
                                                  (short)0, acc01, false, false);
    acc10 = __builtin_amdgcn_wmma_f32_16x16x4_f32(false, x1, false, r0,
                                                  (short)0, acc10, false, false);
    acc11 = __builtin_amdgcn_wmma_f32_16x16x4_f32(false, x1, false, r1,
                                                  (short)0, acc11, false, false);
  }

  // Store 4 tiles; per VGPR row, lanes 0-15 / 16-31 each write a contiguous
  // 64B segment.
#pragma unroll
  for (int r = 0; r < 2; ++r) {
#pragma unroll
    for (int c = 0; c < 2; ++c) {
      const v8f acc = (r == 0) ? (c == 0 ? acc00 : acc01)
                               : (c == 0 ? acc10 : acc11);
      float* op = out + (size_t)(base_m + r * 16 + hi * 8) * dout
                      + (size_t)(base_n + c * 16) + col;
#pragma unroll
      for (int v = 0; v < 8; ++v) {
        float val = acc[v];
        if (RELU) val = fmaxf(val, 0.0f);
        op[(size_t)v * dout] = val;
      }
    }
  }
}

// ---------------------------------------------------------------------------
// Launch: 3 layers. h1/h2/agg live in d_ws (~167 MB total).
// Inputs (setup_inputs order):
//  0:x 1:src0 2:dst0 3:src1 4:dst1 5:src2 6:dst2
//  7:Wl0 8:b0 9:Wr0 10:Wl1 11:b1 12:Wr1 13:Wl2 14:b2 15:Wr2
// ---------------------------------------------------------------------------
extern "C" void kernel_launch(void* const* d_in, const int* in_sizes, int n_in,
                              void* d_out, int out_size, void* d_ws, size_t ws_size,
                              hipStream_t stream) {
  const float* x   = (const float*)d_in[0];
  const int* src0  = (const int*)d_in[1];
  const int* src1  = (const int*)d_in[3];
  const int* src2  = (const int*)d_in[5];
  const float* Wl0 = (const float*)d_in[7];
  const float* b0  = (const float*)d_in[8];
  const float* Wr0 = (const float*)d_in[9];
  const float* Wl1 = (const float*)d_in[10];
  const float* b1  = (const float*)d_in[11];
  const float* Wr1 = (const float*)d_in[12];
  const float* Wl2 = (const float*)d_in[13];
  const float* b2  = (const float*)d_in[14];
  const float* Wr2 = (const float*)d_in[15];
  float* out = (float*)d_out;

  char* ws = (char*)d_ws;
  float* h1  = (float*)(ws);                                   // 102400*256 f32
  float* h2  = (float*)(ws + (size_t)102400 * 256 * 4);        // 10240*256 f32
  float* agg = (float*)(ws + (size_t)102400 * 256 * 4
                            + (size_t)10240 * 256 * 4);        // <= 102400*128 f32

  // ----- layer 0: 1024000x128 -> 102400x256, ReLU -----
  // block = 8 waves covering 32 rows x 256 cols; grid = 102400/32 = 3200
  sage_agg<<<102400, 128, 0, stream>>>(x, src0, agg, 128);
  sage_gemm_wmma<1, 128, 8><<<3200, 256, 0, stream>>>(agg, x, Wl0, Wr0, b0, h1,
                                                      256);
  // ----- layer 1: 102400x256 -> 10240x256, ReLU -----
  sage_agg<<<10240, 256, 0, stream>>>(h1, src1, agg, 256);
  sage_gemm_wmma<1, 256, 8><<<320, 256, 0, stream>>>(agg, h1, Wl1, Wr1, b1, h2,
                                                     256);
  // ----- layer 2: 10240x256 -> 1024x64, no ReLU -----
  // block = 2 waves covering 32 rows x 64 cols; grid = 1024/32 = 32
  sage_agg<<<1024, 256, 0, stream>>>(h2, src2, agg, 256);
  sage_gemm_wmma<0, 256, 2><<<32, 64, 0, stream>>>(agg, h2, Wl2, Wr2, b2, out,
                                                   64);
}